// SpatUFGConv_31851477467635
// MI455X (gfx1250) — compile-verified
//
#include <hip/hip_runtime.h>
#include <hip/hip_bf16.h>
#include <cstdint>

typedef __attribute__((ext_vector_type(2))) float v2f;
typedef __attribute__((ext_vector_type(8))) float v8f;

#define FIN  256
#define FOUT 96
#define EPSV 0.1f

// ---------------------------------------------------------------------------
// init: out = 0, deg = 1 (self loop)
// ---------------------------------------------------------------------------
__global__ __launch_bounds__(256) void init_kernel(float* __restrict__ deg,
                                                   float* __restrict__ out,
                                                   int N) {
    int i = blockIdx.x * 256 + threadIdx.x;
    if (i < N * FOUT) out[i] = 0.0f;
    if (i < N)        deg[i] = 1.0f;
}

// deg[row[e]] += 1 per edge
__global__ __launch_bounds__(256) void deg_count_kernel(const int* __restrict__ row,
                                                        float* __restrict__ deg,
                                                        int E) {
    int e = blockIdx.x * 256 + threadIdx.x;
    if (e < E) atomicAdd(&deg[row[e]], 1.0f);
}

// deg -> eps/deg
__global__ __launch_bounds__(256) void eps_kernel(float* __restrict__ deg, int N) {
    int i = blockIdx.x * 256 + threadIdx.x;
    if (i < N) deg[i] = EPSV / deg[i];
}

// ---------------------------------------------------------------------------
// WMMA f32 GEMM:  t[m,n]  = dsc[m] * sum_k x[m,k] * W[k,n]
//                 t2[m,n] = sign * eps[m] * t[m,n]   (diagonal init for SpMM)
// One wave computes a 16x96 stripe: 6 tiles of 16x16, K-loop of 64 x (16x16x4).
// Block = 256 threads = 8 waves = 128 rows. N % 16 == 0 (50000), so waves are
// either fully in range or fully out (EXEC all-1s for WMMA).
// ---------------------------------------------------------------------------
__global__ __launch_bounds__(256) void gemm_scale_kernel(
    const float* __restrict__ x,     // [N, FIN]
    const float* __restrict__ W,     // [FIN, FOUT], this filter
    const float* __restrict__ dsc,   // [N], this filter
    const float* __restrict__ eps,   // [N] = EPS/deg
    float* __restrict__ t,           // [N, FOUT]
    float* __restrict__ t2,          // [N, FOUT]
    float sign, int N)
{
    const int lane = threadIdx.x & 31;
    const int wave = threadIdx.x >> 5;
    const int m0   = blockIdx.x * 128 + wave * 16;
    if (m0 >= N) return;                       // wave-uniform exit

    const int half = lane >> 4;                // 0: K{0,1}, 1: K{2,3}
    const int l16  = lane & 15;

    v8f acc[6] = {};                           // 6 x (16x16 f32) = 16x96

    const float* xrow = x + (size_t)(m0 + l16) * FIN;

    #pragma unroll 4
    for (int k0 = 0; k0 < FIN; k0 += 4) {
        const int ka = k0 + 2 * half;
        // A frag (16x4 f32 layout): lane m=l16 holds x[m][ka], x[m][ka+1]
        v2f a = *(const v2f*)(xrow + ka);
        #pragma unroll
        for (int j = 0; j < 6; ++j) {
            // B frag (4x16 f32 layout): lane n=l16 holds W[ka][n], W[ka+1][n]
            const int n = j * 16 + l16;
            v2f b;
            b.x = W[(size_t)ka       * FOUT + n];
            b.y = W[(size_t)(ka + 1) * FOUT + n];
            acc[j] = __builtin_amdgcn_wmma_f32_16x16x4_f32(
                false, a, false, b, (short)0, acc[j], false, false);
        }
    }

    // C/D layout: VGPR v, lanes 0-15 -> M=v, lanes 16-31 -> M=v+8; N = l16
    #pragma unroll
    for (int v = 0; v < 8; ++v) {
        const int m  = m0 + v + 8 * half;
        const float ds = dsc[m];
        const float se = sign * eps[m];
        #pragma unroll
        for (int j = 0; j < 6; ++j) {
            const int n = j * 16 + l16;
            const float val = acc[j][v] * ds;
            t [(size_t)m * FOUT + n] = val;
            t2[(size_t)m * FOUT + n] = se * val;
        }
    }
}

// ---------------------------------------------------------------------------
// SpMM: t2[row[e], :] += adj[e] * t[col[e], :]   (one wave per edge,
// lanes cover 96 floats in 3 coalesced strides; t/t2 are L2-resident)
// ---------------------------------------------------------------------------
__global__ __launch_bounds__(256) void spmm_kernel(
    const int* __restrict__ row, const int* __restrict__ col,
    const float* __restrict__ adj, const float* __restrict__ t,
    float* __restrict__ t2, int E)
{
    const int wid  = (blockIdx.x * 256 + threadIdx.x) >> 5;  // edge index
    const int lane = threadIdx.x & 31;
    if (wid >= E) return;
    const int   r = row[wid];
    const int   c = col[wid];
    const float a = adj[wid];
    const float* __restrict__ src = t  + (size_t)c * FOUT;
    float*       __restrict__ dst = t2 + (size_t)r * FOUT;
    #pragma unroll
    for (int f = lane; f < FOUT; f += 32)
        atomicAdd(&dst[f], a * src[f]);
}

// ---------------------------------------------------------------------------
// out += dsc * relu(t2)  (+ bias on the last filter)
// ---------------------------------------------------------------------------
__global__ __launch_bounds__(256) void finalize_kernel(
    const float* __restrict__ t2, const float* __restrict__ dsc,
    const float* __restrict__ bias, float* __restrict__ out,
    int N, int addBias)
{
    int idx = blockIdx.x * 256 + threadIdx.x;
    if (idx >= N * FOUT) return;
    const int n = idx / FOUT;
    const int f = idx - n * FOUT;
    float v = t2[idx];
    v = v > 0.0f ? v : 0.0f;
    float r = out[idx] + dsc[n] * v;
    if (addBias) r += bias[f];
    out[idx] = r;
}

// ---------------------------------------------------------------------------
extern "C" void kernel_launch(void* const* d_in, const int* in_sizes, int n_in,
                              void* d_out, int out_size, void* d_ws, size_t ws_size,
                              hipStream_t stream) {
    const float* x    = (const float*)d_in[0];   // [N, 256]
    const float* adj  = (const float*)d_in[1];   // [E]
    const float* dsc  = (const float*)d_in[2];   // [NF, N]
    const float* W    = (const float*)d_in[3];   // [NF, 256, 96]
    const float* bias = (const float*)d_in[4];   // [96]
    const int*   eidx = (const int*)  d_in[5];   // [2, E]

    const int N  = in_sizes[0] / FIN;
    const int E  = in_sizes[1];
    const int NF = in_sizes[2] / N;

    const int* row = eidx;
    const int* col = eidx + E;
    float* out = (float*)d_out;

    // workspace: deg/eps [N] | t [N*96] | t2 [N*96]
    float* deg = (float*)d_ws;
    float* t   = deg + (((size_t)N + 63) & ~(size_t)63);
    float* t2  = t + (size_t)N * FOUT;

    const int totalNF = N * FOUT;

    init_kernel     <<<(totalNF + 255) / 256, 256, 0, stream>>>(deg, out, N);
    deg_count_kernel<<<(E + 255) / 256,       256, 0, stream>>>(row, deg, E);
    eps_kernel      <<<(N + 255) / 256,       256, 0, stream>>>(deg, N);

    for (int ii = 0; ii < NF; ++ii) {
        const float sign = (ii == 0) ? -1.0f : 1.0f;
        gemm_scale_kernel<<<(N + 127) / 128, 256, 0, stream>>>(
            x, W + (size_t)ii * FIN * FOUT, dsc + (size_t)ii * N,
            deg, t, t2, sign, N);
        spmm_kernel<<<(E + 7) / 8, 256, 0, stream>>>(row, col, adj, t, t2, E);
        finalize_kernel<<<(totalNF + 255) / 256, 256, 0, stream>>>(
            t2, dsc + (size_t)ii * N, bias, out, N, ii == NF - 1);
    }
}